// CarafeConvolution_47364899340485
// MI455X (gfx1250) — compile-verified
//
#include <hip/hip_runtime.h>
#include <math.h>

typedef __attribute__((ext_vector_type(2))) float v2f;
typedef __attribute__((ext_vector_type(4))) float v4f;
typedef __attribute__((ext_vector_type(8))) float v8f;

#define BATCH 2
#define HIN   64
#define WIN   64
#define CH    256
#define HOUT  128
#define WOUT  128
#define KK2   25          // 5x5 taps
#define FROW  260         // padded LDS row stride for 6x6 patch (256 + 4)
#define OROW  260         // padded LDS row stride for output tile
#define BROW  102         // Bm row stride: >=100, 8B-pair aligned, good banking

// CARAFE as a block-diagonal GEMM on V_WMMA_F32_16X16X4_F32.
// One workgroup = one 2x2 low-res block = 4x4 high-res output pixels x 256 ch.
// D[16ch,16outpix] += A[16ch,K] * B[K,16outpix], K = 100 (4 patches x 25 taps),
// 25 WMMA steps of K=4, two channel tiles per wave fused on a shared B fragment.
__global__ __launch_bounds__(256) void carafe_wmma_kernel(
    const float* __restrict__ feat_g,   // [2,64,64,256]
    const float* __restrict__ mask_g,   // [2,128,128,25]
    float* __restrict__ out_g)          // [2,128,128,256]
{
    __shared__ float featl[36 * FROW];      // 6x6 patch union, 37440 B
    __shared__ float Bm[16][BROW];          // dense block-diag masks, 6528 B
    __shared__ float outl[16][OROW];        // output tile staging, 16640 B
    __shared__ int   adrtab[100];           // K-index -> featl row offset

    const int tid = threadIdx.x;
    int bid = blockIdx.x;
    const int lw = bid & 31; bid >>= 5;     // low-res block x (of 32)
    const int lh = bid & 31; bid >>= 5;     // low-res block y (of 32)
    const int bz = bid;                     // batch
    const int h0 = lh * 2, w0 = lw * 2;     // low-res origin of 2x2 block

    // ---- K-decode table: kk = q*25+p -> LDS row offset of feature tap -----
    if (tid < 100) {
        int q = tid / KK2;
        int p = tid - q * KK2;
        int y = (q >> 1) + p / 5;
        int x = (q & 1) + p % 5;
        adrtab[tid] = (y * 6 + x) * FROW;
    }

    // ---- zero the B matrix (only nonzeros are scattered later) -----------
    for (int it = 0; it < 7; ++it) {
        int idx = it * 256 + tid;           // 16*BROW = 1632 floats
        if (idx < 16 * BROW) ((float*)Bm)[idx] = 0.f;
    }

    // ---- stage 6x6x256 feature patch union (zero-padded) as float4 -------
    for (int it = 0; it < 9; ++it) {
        int idx = it * 256 + tid;           // 2304 float4 total
        int cv  = idx & 63;
        int xy  = idx >> 6;
        int x = xy % 6, y = xy / 6;
        int gy = h0 - 2 + y, gx = w0 - 2 + x;
        v4f val = {0.f, 0.f, 0.f, 0.f};
        if (gy >= 0 && gy < HIN && gx >= 0 && gx < WIN) {
            val = *(const v4f*)(feat_g +
                  (((size_t)bz * HIN + gy) * WIN + gx) * CH + cv * 4);
        }
        *(v4f*)(&featl[(y * 6 + x) * FROW + cv * 4]) = val;
    }

    // ---- softmax weights into registers (global reads only, pre-sync) ----
    float w[KK2];
    int myq_s = 0;
    if (tid < 16) {
        int s = tid;
        myq_s = s >> 2;
        int t = s & 3;
        int Y = (h0 + (myq_s >> 1)) * 2 + (t >> 1);
        int X = (w0 + (myq_s & 1)) * 2 + (t & 1);
        const float* mp = mask_g + (((size_t)bz * HOUT + Y) * WOUT + X) * KK2;
        float mx = mp[0];
        #pragma unroll
        for (int p = 1; p < KK2; ++p) mx = fmaxf(mx, mp[p]);
        float sum = 0.f;
        #pragma unroll
        for (int p = 0; p < KK2; ++p) { w[p] = __expf(mp[p] - mx); sum += w[p]; }
        float inv = 1.f / sum;
        #pragma unroll
        for (int p = 0; p < KK2; ++p) w[p] *= inv;
    }
    __syncthreads();   // zero-fill + feature staging complete

    // ---- scatter nonzero block-diagonal entries: Bm[s][q*25+p] -----------
    if (tid < 16) {
        #pragma unroll
        for (int p = 0; p < KK2; ++p)
            Bm[tid][myq_s * KK2 + p] = w[p];
    }
    __syncthreads();

    // ---- WMMA phase: 8 waves, 2 channel tiles each, shared B fragment ----
    const int wave = tid >> 5;
    const int lane = tid & 31;
    const int n16  = lane & 15;             // A row M / B col N / D col N
    const int hi   = lane >> 4;             // K sub-offset (+2) & D row (+8)
    const int c0a  = wave * 16 + n16;       // channel tile 0 lane channel
    const int c0b  = (wave + 8) * 16 + n16; // channel tile 1 lane channel

    v8f acc0 = {0.f, 0.f, 0.f, 0.f, 0.f, 0.f, 0.f, 0.f};
    v8f acc1 = {0.f, 0.f, 0.f, 0.f, 0.f, 0.f, 0.f, 0.f};

    #pragma unroll
    for (int k4 = 0; k4 < 25; ++k4) {       // K = 100 in steps of 4
        int kk0 = k4 * 4 + hi * 2;
        int o0 = adrtab[kk0];
        int o1 = adrtab[kk0 + 1];

        v2f av0, av1, bv;
        av0.x = featl[o0 + c0a];
        av0.y = featl[o1 + c0a];
        av1.x = featl[o0 + c0b];
        av1.y = featl[o1 + c0b];
        bv = *(const v2f*)(&Bm[n16][kk0]);  // contiguous 8B, unconditional

        acc0 = __builtin_amdgcn_wmma_f32_16x16x4_f32(
            false, av0, false, bv, (short)0, acc0, false, false);
        acc1 = __builtin_amdgcn_wmma_f32_16x16x4_f32(
            false, av1, false, bv, (short)0, acc1, false, false);
    }

    // D layout: VGPR r -> M = hi*8 + r, N = n16
    #pragma unroll
    for (int r = 0; r < 8; ++r) {
        outl[n16][wave * 16 + hi * 8 + r]       = acc0[r];
        outl[n16][(wave + 8) * 16 + hi * 8 + r] = acc1[r];
    }
    __syncthreads();

    // ---- coalesced output: each outpix row is 1 KB contiguous in global ---
    for (int it = 0; it < 4; ++it) {
        int idx = it * 256 + tid;           // 1024 float4 total
        int s  = idx >> 6;
        int cv = idx & 63;
        int q = s >> 2, t = s & 3;
        int Y = (h0 + (q >> 1)) * 2 + (t >> 1);
        int X = (w0 + (q & 1)) * 2 + (t & 1);
        v4f val = *(v4f*)(&outl[s][cv * 4]);
        *(v4f*)(out_g + (((size_t)bz * HOUT + Y) * WOUT + X) * CH + cv * 4) = val;
    }
}

extern "C" void kernel_launch(void* const* d_in, const int* in_sizes, int n_in,
                              void* d_out, int out_size, void* d_ws, size_t ws_size,
                              hipStream_t stream) {
    const float* features = (const float*)d_in[0];
    const float* masks    = (const float*)d_in[1];
    float* out            = (float*)d_out;
    dim3 grid(BATCH * 32 * 32);     // one block per 2x2 low-res block
    carafe_wmma_kernel<<<grid, 256, 0, stream>>>(features, masks, out);
}